// DeltaSynapse_49091476193684
// MI455X (gfx1250) — compile-verified
//
#include <hip/hip_runtime.h>

// I[b,o] = sum_{d,e} (signs*W)[e,o] * Xd[d,b,e] * delaymap[d,e,o] * Wshort[d,b,e]
// Treated as a 16 x 4096 x (8*4096) f32 GEMM on V_WMMA_F32_16X16X4_F32.
// Memory-bound: delaymap (512MB) + W/signs (128MB) streamed once, NT-hinted.

typedef __attribute__((ext_vector_type(2))) float v2f;
typedef __attribute__((ext_vector_type(8))) float v8f;

constexpr int D = 8, B = 16, E = 4096, O = 4096;
constexpr int KSPLIT = 32;                 // e-chunks (partial sums via f32 atomics)
constexpr int EC = E / KSPLIT;             // 128 e-values per chunk
constexpr int OT = 4;                      // 16-wide o-tiles per wave (64 cols)
constexpr int WPB = 4;                     // waves per block
constexpr int OCOLS_PER_BLOCK = WPB * OT * 16;  // 256

__global__ void zero_kernel(float* __restrict__ p, int nElems) {
  int i = blockIdx.x * blockDim.x + threadIdx.x;
  if (i < nElems) p[i] = 0.0f;
}

__global__ __launch_bounds__(WPB * 32)
void delta_synapse_wmma(const float* __restrict__ W,
                        const float* __restrict__ signs,
                        const float* __restrict__ Xd,
                        const float* __restrict__ dmap,
                        const float* __restrict__ Wshort,
                        float* __restrict__ out) {
  const int lane = threadIdx.x & 31;
  const int wave = threadIdx.x >> 5;
  const int n    = lane & 15;        // B/C/D fragment column, A fragment row (M)
  const int half = lane >> 4;        // 0: K=ka in v0; 1: K=ka in v0 with ka=2
  const int ka   = half << 1;        // K offset held in fragment VGPR0 (VGPR1 = ka+1)
  const int oBase  = blockIdx.x * OCOLS_PER_BLOCK + wave * (OT * 16);
  const int eStart = blockIdx.y * EC;

  v8f c[OT];
#pragma unroll
  for (int t = 0; t < OT; ++t) c[t] = v8f{};

  for (int et = 0; et < EC; et += 4) {
    const int  e0  = eStart + et;
    const long rw0 = (long)(e0 + ka) * O;   // row e0+ka   of (E,O) arrays
    const long rw1 = rw0 + O;               // row e0+ka+1

    // Weff = signs*W for this K-tile, reused across all 8 delays.
    float w0[OT], w1[OT];
#pragma unroll
    for (int t = 0; t < OT; ++t) {
      const long o = oBase + t * 16 + n;
      w0[t] = __builtin_nontemporal_load(W + rw0 + o) *
              __builtin_nontemporal_load(signs + rw0 + o);
      w1[t] = __builtin_nontemporal_load(W + rw1 + o) *
              __builtin_nontemporal_load(signs + rw1 + o);
    }

#pragma unroll
    for (int d = 0; d < D; ++d) {
      // A fragment: A[d, b=n, e0+ka .. e0+ka+1] = Xd*Wshort (L2-resident, float2 load)
      const long aIdx = ((long)d * B + n) * E + e0 + ka;
      const v2f x  = *(const v2f*)(Xd + aIdx);
      const v2f sh = *(const v2f*)(Wshort + aIdx);
      v2f a;
      a.x = x.x * sh.x;
      a.y = x.y * sh.y;

      const long dm0 = (long)d * E * O + rw0;  // delaymap rows for this (d, K-tile)
      const long dm1 = dm0 + O;

#pragma unroll
      for (int t = 0; t < OT; ++t) {           // A fragment reused across 4 o-tiles
        const long o = oBase + t * 16 + n;
        v2f bf;
        bf.x = w0[t] * __builtin_nontemporal_load(dmap + dm0 + o);
        bf.y = w1[t] * __builtin_nontemporal_load(dmap + dm1 + o);
        c[t] = __builtin_amdgcn_wmma_f32_16x16x4_f32(
            false, a, false, bf, (short)0, c[t], false, false);
      }
    }
  }

  // C/D layout: lane holds rows M = r + half*8, col N = n. Accumulate partials.
#pragma unroll
  for (int t = 0; t < OT; ++t) {
    const int o = oBase + t * 16 + n;
#pragma unroll
    for (int r = 0; r < 8; ++r) {
      const int brow = r + half * 8;
      unsafeAtomicAdd(out + (long)brow * O + o, c[t][r]);
    }
  }
}

extern "C" void kernel_launch(void* const* d_in, const int* in_sizes, int n_in,
                              void* d_out, int out_size, void* d_ws, size_t ws_size,
                              hipStream_t stream) {
  const float* W      = (const float*)d_in[0];  // (E,O)
  const float* signs  = (const float*)d_in[1];  // (E,O)
  const float* Xd     = (const float*)d_in[2];  // (D,B,E)
  const float* dmap   = (const float*)d_in[3];  // (D,E,O)
  const float* Wshort = (const float*)d_in[4];  // (D,B,E)
  float* out = (float*)d_out;                   // (B,O)

  zero_kernel<<<(B * O + 255) / 256, 256, 0, stream>>>(out, B * O);

  dim3 grid(O / OCOLS_PER_BLOCK, KSPLIT);       // (16, 32)
  delta_synapse_wmma<<<grid, WPB * 32, 0, stream>>>(W, signs, Xd, dmap, Wshort, out);
}